// MahalanobisDistanceConstraint_24927990186059
// MI455X (gfx1250) — compile-verified
//
#include <hip/hip_runtime.h>
#include <math.h>

#define K_NEI       8
#define N_PTS       8192
#define M_GAUSS     32768
#define MAX_DIST2   (0.05f * 0.05f)
#define SURF_TOL    0.002f
#define PEN_PENALTY 10.0f

#define G_SPLIT     4                    // gaussian-dimension split (parallelism)
#define M_PER_SPLIT (M_GAUSS / G_SPLIT)  // 8192 gaussians per block
#define SUPER       512                  // gaussians staged in LDS per step
#define PTS_PER_BLK 128                  // 8 waves * 16 points

typedef __attribute__((ext_vector_type(2))) float v2f;
typedef __attribute__((ext_vector_type(8))) float v8f;

// ---------------------------------------------------------------------------
// Phase 0: zero the scalar accumulator (harness poisons d_out and does not
// re-poison between graph replays).
// ---------------------------------------------------------------------------
__global__ void init_out_kernel(float* __restrict__ out) {
    out[0] = 0.0f;
}

// ---------------------------------------------------------------------------
// Phase 1: partial kNN via V_WMMA_F32_16X16X4_F32.
//
//   A (16x4, M=gaussian rows):  [gx, gy, gz, ||g||^2]
//   B (4x16, N=point columns):  [-2px, -2py, -2pz, 1]^T
//   D[m][n] = ||g_m||^2 - 2 p_n.g_m  == d2 minus the row-constant ||p||^2
//
// C/D layout (wave32): VGPR i -> row m=i (lanes 0-15) / m=i+8 (lanes 16-31),
// column n = lane&15 => each lane owns ONE contact point, 8 gaussians/WMMA;
// lanes L and L+16 cover disjoint rows of the same point (merged via shfl).
//
// Grid: (N/128, G_SPLIT). A block = 8 waves = 128 points, scans M/4 gaussians
// staged through LDS in 512-gaussian super-chunks (16B/gaussian, norm^2
// precomputed at fill). Writes a partial top-8 per (split, point).
// ---------------------------------------------------------------------------
__global__ __launch_bounds__(256) void knn_phase1_kernel(
    const float* __restrict__ pts,     // [N,3]
    const float* __restrict__ gpos,    // [M,3]
    float*       __restrict__ ws_d2,   // [G_SPLIT,N,8]
    int*         __restrict__ ws_idx)  // [G_SPLIT,N,8]
{
    __shared__ float sG[SUPER * 4];    // gx,gy,gz,||g||^2 per staged gaussian

    const int  lane = threadIdx.x & 31;
    const int  wIB  = threadIdx.x >> 5;         // wave in block (0..7)
    const int  col  = lane & 15;                // point column within tile
    const bool hi   = lane >= 16;               // upper half: K=2,3 / rows m+8
    const int  pt   = blockIdx.x * PTS_PER_BLK + wIB * 16 + col;
    const int  gBase = blockIdx.y * M_PER_SPLIT;

    // Contact point (wave-invariant over the whole scan).
    const float px = pts[pt * 3 + 0];
    const float py = pts[pt * 3 + 1];
    const float pz = pts[pt * 3 + 2];
    const float p2 = px * px + py * py + pz * pz;

    // B (4x16): VGPR0 = K0 (lanes 0-15) / K2 (lanes 16-31); VGPR1 = K1 / K3.
    v2f b;
    b.x = hi ? (-2.0f * pz) : (-2.0f * px);
    b.y = hi ? 1.0f         : (-2.0f * py);

    // Per-lane ascending top-8 (value = ||g||^2 - 2 p.g, order == d2 order).
    float sv[K_NEI];
    int   si[K_NEI];
#pragma unroll
    for (int j = 0; j < K_NEI; ++j) { sv[j] = 3.0e38f; si[j] = 0; }

    for (int sc = 0; sc < M_PER_SPLIT / SUPER; ++sc) {
        // ---- cooperative LDS fill: 512 gaussians, 2 per thread ----
        __syncthreads();
#pragma unroll
        for (int r = 0; r < SUPER / 256; ++r) {
            const int j  = threadIdx.x + r * 256;          // staged slot
            const int gg = gBase + sc * SUPER + j;         // global gaussian
            const float gx = gpos[gg * 3 + 0];
            const float gy = gpos[gg * 3 + 1];
            const float gz = gpos[gg * 3 + 2];
            sG[j * 4 + 0] = gx;
            sG[j * 4 + 1] = gy;
            sG[j * 4 + 2] = gz;
            sG[j * 4 + 3] = gx * gx + gy * gy + gz * gz;
        }
        __syncthreads();

        // ---- 32 WMMA chunks of 16 gaussians from LDS ----
#pragma unroll 2
        for (int c = 0; c < SUPER / 16; ++c) {
            const int sl = c * 16 + col;                   // staged slot
            const float gx = sG[sl * 4 + 0];
            const float gy = sG[sl * 4 + 1];
            const float gz = sG[sl * 4 + 2];
            const float n2 = sG[sl * 4 + 3];

            v2f a;                                         // A (16x4)
            a.x = hi ? gz : gx;
            a.y = hi ? n2 : gy;

            v8f d = {};
            d = __builtin_amdgcn_wmma_f32_16x16x4_f32(
                    /*neg_a=*/false, a, /*neg_b=*/false, b,
                    /*c_mod=*/(short)0, d, /*reuse_a=*/false, /*reuse_b=*/false);

            // Fast reject: skip insertion unless a candidate beats the 8th.
            const float m01 = fminf(d[0], d[1]), m23 = fminf(d[2], d[3]);
            const float m45 = fminf(d[4], d[5]), m67 = fminf(d[6], d[7]);
            const float cmin = fminf(fminf(m01, m23), fminf(m45, m67));
            if (cmin < sv[K_NEI - 1]) {
#pragma unroll
                for (int i = 0; i < 8; ++i) {
                    const float v = d[i];
                    if (v < sv[K_NEI - 1]) {
                        float cv = v;
                        int   ci = gBase + sc * SUPER + c * 16 + i + (hi ? 8 : 0);
#pragma unroll
                        for (int j = 0; j < K_NEI; ++j) {
                            const bool  lt = cv < sv[j];
                            const float ns = lt ? cv    : sv[j];
                            const int   ni = lt ? ci    : si[j];
                            const float nc = lt ? sv[j] : cv;
                            const int   nj = lt ? si[j] : ci;
                            sv[j] = ns; si[j] = ni; cv = nc; ci = nj;
                        }
                    }
                }
            }
        }
    }

    // Merge the two half-wave streams of the same contact point.
#pragma unroll
    for (int j = 0; j < K_NEI; ++j) {
        const float ov = __shfl_xor(sv[j], 16, 32);
        const int   oi = __shfl_xor(si[j], 16, 32);
        if (ov < sv[K_NEI - 1]) {
            float cv = ov; int ci = oi;
#pragma unroll
            for (int k = 0; k < K_NEI; ++k) {
                const bool  lt = cv < sv[k];
                const float ns = lt ? cv    : sv[k];
                const int   ni = lt ? ci    : si[k];
                const float nc = lt ? sv[k] : cv;
                const int   nj = lt ? si[k] : ci;
                sv[k] = ns; si[k] = ni; cv = nc; ci = nj;
            }
        }
    }

    if (!hi) {
        const int base = (blockIdx.y * N_PTS + pt) * K_NEI;
#pragma unroll
        for (int j = 0; j < K_NEI; ++j) {
            ws_d2[base + j]  = sv[j] + p2;     // true squared distance
            ws_idx[base + j] = si[j];
        }
    }
}

// ---------------------------------------------------------------------------
// Phase 2: merge the G_SPLIT partial lists into the exact top-8, then compute
// the Mahalanobis surface distance + loss reduction. One thread per point.
// ---------------------------------------------------------------------------
__global__ __launch_bounds__(256) void mahal_phase2_kernel(
    const float* __restrict__ pts,     // [N,3]
    const float* __restrict__ gpos,    // [M,3]
    const float* __restrict__ grot,    // [M,4] (w,x,y,z)
    const float* __restrict__ gscl,    // [M,3] log-sigma
    const float* __restrict__ cw,      // [N]
    const float* __restrict__ ws_d2,   // [G_SPLIT,N,8]
    const int*   __restrict__ ws_idx,  // [G_SPLIT,N,8]
    float*       __restrict__ out)     // [1]
{
    const int n = blockIdx.x * blockDim.x + threadIdx.x;
    float r2 = 0.0f;
    if (n < N_PTS) {
        // ---- exact top-8 from 4 partial top-8 lists ----
        float sv[K_NEI];
        int   si[K_NEI];
#pragma unroll
        for (int j = 0; j < K_NEI; ++j) { sv[j] = 3.0e38f; si[j] = 0; }
#pragma unroll
        for (int s = 0; s < G_SPLIT; ++s) {
            const int base = (s * N_PTS + n) * K_NEI;
#pragma unroll
            for (int j = 0; j < K_NEI; ++j) {
                const float v  = ws_d2[base + j];
                const int   id = ws_idx[base + j];
                if (v < sv[K_NEI - 1]) {
                    float cv = v; int ci = id;
#pragma unroll
                    for (int k = 0; k < K_NEI; ++k) {
                        const bool  lt = cv < sv[k];
                        const float ns = lt ? cv    : sv[k];
                        const int   ni = lt ? ci    : si[k];
                        const float nc = lt ? sv[k] : cv;
                        const int   nj = lt ? si[k] : ci;
                        sv[k] = ns; si[k] = ni; cv = nc; ci = nj;
                    }
                }
            }
        }

        const float px = pts[n * 3 + 0];
        const float py = pts[n * 3 + 1];
        const float pz = pts[n * 3 + 2];

        float surface = 1000.0f;
#pragma unroll 1
        for (int k = 0; k < K_NEI; ++k) {
            const int   g  = si[k];
            const float d2 = sv[k];

            // Normalized quaternion -> rotation matrix.
            float qw = grot[g * 4 + 0], qx = grot[g * 4 + 1];
            float qy = grot[g * 4 + 2], qz = grot[g * 4 + 3];
            const float qn  = sqrtf(qw * qw + qx * qx + qy * qy + qz * qz);
            const float inv = 1.0f / fmaxf(qn, 1e-8f);
            qw *= inv; qx *= inv; qy *= inv; qz *= inv;
            const float xx = qx * qx, yy = qy * qy, zz = qz * qz;
            const float xy = qx * qy, xz = qx * qz, yz = qy * qz;
            const float wx = qw * qx, wy = qw * qy, wz = qw * qz;
            const float r00 = 1.0f - 2.0f * (yy + zz), r01 = 2.0f * (xy - wz), r02 = 2.0f * (xz + wy);
            const float r10 = 2.0f * (xy + wz), r11 = 1.0f - 2.0f * (xx + zz), r12 = 2.0f * (yz - wx);
            const float r20 = 2.0f * (xz - wy), r21 = 2.0f * (yz + wx), r22 = 1.0f - 2.0f * (xx + yy);

            // sigma^2 with the reference's clamping.
            float s0 = fminf(fmaxf(gscl[g * 3 + 0], -5.0f), 5.0f);
            float s1 = fminf(fmaxf(gscl[g * 3 + 1], -5.0f), 5.0f);
            float s2 = fminf(fmaxf(gscl[g * 3 + 2], -5.0f), 5.0f);
            float v0 = fminf(fmaxf(expf(s0), 1e-4f), 1.0f); v0 *= v0;
            float v1 = fminf(fmaxf(expf(s1), 1e-4f), 1.0f); v1 *= v1;
            float v2 = fminf(fmaxf(expf(s2), 1e-4f), 1.0f); v2 *= v2;

            // cov = R diag(sigma^2) R^T + 1e-6 I  (symmetric, 6 entries).
            const float c00 = r00 * r00 * v0 + r01 * r01 * v1 + r02 * r02 * v2 + 1e-6f;
            const float c01 = r00 * r10 * v0 + r01 * r11 * v1 + r02 * r12 * v2;
            const float c02 = r00 * r20 * v0 + r01 * r21 * v1 + r02 * r22 * v2;
            const float c11 = r10 * r10 * v0 + r11 * r11 * v1 + r12 * r12 * v2 + 1e-6f;
            const float c12 = r10 * r20 * v0 + r11 * r21 * v1 + r12 * r22 * v2;
            const float c22 = r20 * r20 * v0 + r21 * r21 * v1 + r22 * r22 * v2 + 1e-6f;

            // SPD 3x3 inverse via adjugate (== Cholesky solve result).
            const float M00 = c11 * c22 - c12 * c12;
            const float M01 = c02 * c12 - c01 * c22;
            const float M02 = c01 * c12 - c02 * c11;
            const float det = c00 * M00 + c01 * M01 + c02 * M02;
            const float rd  = 1.0f / det;
            const float i00 = M00 * rd, i01 = M01 * rd, i02 = M02 * rd;
            const float i11 = (c00 * c22 - c02 * c02) * rd;
            const float i12 = (c02 * c01 - c00 * c12) * rd;
            const float i22 = (c00 * c11 - c01 * c01) * rd;

            const float dx = px - gpos[g * 3 + 0];
            const float dy = py - gpos[g * 3 + 1];
            const float dz = pz - gpos[g * 3 + 2];
            float quad = i00 * dx * dx + i11 * dy * dy + i22 * dz * dz
                       + 2.0f * (i01 * dx * dy + i02 * dx * dz + i12 * dy * dz);
            if (quad != quad) quad = 0.0f;                 // nan -> 0
            quad = fminf(fmaxf(quad, 0.0f), 1e6f);         // -inf->0, +inf->1e6
            const float mahal = sqrtf(quad);

            // valid = sqrt(max(d2,0)) < 0.05  <=>  d2 < 0.0025
            const float masked = (d2 < MAX_DIST2) ? mahal : 1000.0f;
            surface = fminf(surface, masked);
        }
        if (surface != surface) surface = 1000.0f;

        float res = fmaxf(surface - SURF_TOL, 0.0f) * cw[n];
        if (surface < 0.0f) res *= PEN_PENALTY;
        r2 = res * res * (1.0f / (float)N_PTS);
    }

    // wave32 reduction, then one f32 atomic per wave.
#pragma unroll
    for (int o = 16; o > 0; o >>= 1) r2 += __shfl_xor(r2, o, 32);
    if ((threadIdx.x & 31) == 0) atomicAdd(out, r2);
}

// ---------------------------------------------------------------------------
extern "C" void kernel_launch(void* const* d_in, const int* in_sizes, int n_in,
                              void* d_out, int out_size, void* d_ws, size_t ws_size,
                              hipStream_t stream) {
    (void)in_sizes; (void)n_in; (void)out_size; (void)ws_size;

    const float* pts  = (const float*)d_in[0];   // [8192,3]
    const float* gpos = (const float*)d_in[1];   // [32768,3]
    const float* grot = (const float*)d_in[2];   // [32768,4]
    const float* gscl = (const float*)d_in[3];   // [32768,3]
    const float* cw   = (const float*)d_in[4];   // [8192]
    float* out = (float*)d_out;

    // Workspace: [G_SPLIT*N*8 f32][G_SPLIT*N*8 i32] = 2 MB.
    float* ws_d2  = (float*)d_ws;
    int*   ws_idx = (int*)((char*)d_ws +
                           (size_t)G_SPLIT * N_PTS * K_NEI * sizeof(float));

    init_out_kernel<<<1, 1, 0, stream>>>(out);

    // 64 point-tiles x 4 gaussian splits = 256 blocks of 8 waves = 2048 waves.
    dim3 grid1(N_PTS / PTS_PER_BLK, G_SPLIT);
    knn_phase1_kernel<<<grid1, 256, 0, stream>>>(pts, gpos, ws_d2, ws_idx);

    mahal_phase2_kernel<<<(N_PTS + 255) / 256, 256, 0, stream>>>(
        pts, gpos, grot, gscl, cw, ws_d2, ws_idx, out);
}